// RGCN_LP_47811575939204
// MI455X (gfx1250) — compile-verified
//
#include <hip/hip_runtime.h>

// MI455X / gfx1250: wave32, WMMA fp32 path (V_WMMA_F32_16X16X4_F32).
// Pipeline is HBM/atomic-bound (~3 GB traffic, ~25 GF), so full-precision
// f32 WMMA is free and keeps accuracy through 3 convs + LayerNorm.

typedef __attribute__((ext_vector_type(2))) float v2f;
typedef __attribute__((ext_vector_type(8))) float v8f;

#define DCH 128
#define NREL 4
#define NBASES 4

// ---------------------------------------------------------------- utilities

__global__ void fill_kernel(float* __restrict__ p, int n, float v) {
  int i = blockIdx.x * blockDim.x + threadIdx.x;
  if (i < n) p[i] = v;
}

__global__ void relu_kernel(float* __restrict__ x, int n) {
  int i = blockIdx.x * blockDim.x + threadIdx.x;
  if (i < n) x[i] = fmaxf(x[i], 0.0f);
}

// W[r] = sum_b comp[r,b] * bases[b]  (4 x 128 x 128 output)
__global__ void combine_weights_kernel(const float* __restrict__ comp,
                                       const float* __restrict__ bases,
                                       float* __restrict__ W) {
  int idx = blockIdx.x * blockDim.x + threadIdx.x;  // [0, 4*128*128)
  int r = idx >> 14;
  int io = idx & 16383;
  float acc = 0.0f;
#pragma unroll
  for (int b = 0; b < NBASES; ++b)
    acc += comp[r * NBASES + b] * bases[b * (DCH * DCH) + io];
  W[idx] = acc;
}

// cnt[dst*4 + rel] += 1  (rel = edge_type // 2)
__global__ void count_kernel(const int* __restrict__ ei, const int* __restrict__ et,
                             float* __restrict__ cnt, int E) {
  int e = blockIdx.x * blockDim.x + threadIdx.x;
  if (e >= E) return;
  int dst = ei[E + e];
  int r = et[e] >> 1;
  atomicAdd(&cnt[(size_t)dst * NREL + r], 1.0f);
}

// in-place: c = 1 / max(c, 1)
__global__ void invcnt_kernel(float* __restrict__ c, int n) {
  int i = blockIdx.x * blockDim.x + threadIdx.x;
  if (i < n) c[i] = 1.0f / fmaxf(c[i], 1.0f);
}

// agg[dst,:] += x[src,:] for edges of relation r. One wave32 per edge,
// float4 gather per lane, 4 fp32 global atomics per lane.
__global__ void aggregate_kernel(const float* __restrict__ x,
                                 const int* __restrict__ ei,
                                 const int* __restrict__ et,
                                 int r, float* __restrict__ agg, int E) {
  int tid = blockIdx.x * blockDim.x + threadIdx.x;
  int e = tid >> 5;
  if (e >= E) return;
  if ((et[e] >> 1) != r) return;
  int lane = tid & 31;
  int s = ei[e];
  int dst = ei[E + e];
  const float4 v = ((const float4*)(x + (size_t)s * DCH))[lane];
  float* o = agg + (size_t)dst * DCH + lane * 4;
  atomicAdd(o + 0, v.x);
  atomicAdd(o + 1, v.y);
  atomicAdd(o + 2, v.z);
  atomicAdd(o + 3, v.w);
}

// ------------------------------------------------------------- WMMA GEMM
// C[M,128] (+)= (rowScale * A[M,K]) @ B   using V_WMMA_F32_16X16X4_F32.
// TRANSB: B stored [128,K] row-major (i.e. compute A @ B^T), else B is [K,128].
// One wave per 16x16 tile; block = 8 waves covering 16 rows x all 128 cols.
template <bool TRANSB, bool ACCUM>
__global__ __launch_bounds__(256) void wmma_gemm_kernel(
    const float* __restrict__ A, const float* __restrict__ B,
    float* __restrict__ C, const float* __restrict__ bias,
    const float* __restrict__ rowScale,  // pre-offset by relation, stride NREL
    int M, int K) {
  const int lane = threadIdx.x & 31;
  const int wave = threadIdx.x >> 5;
  const int mbase = blockIdx.x * 16;
  const int nbase = wave * 16;
  const int mr = lane & 15;   // A row within tile / B-C column within tile
  const int half = lane >> 4; // selects K pair (ISA 7.12.2 f32 layouts)

  int arow = mbase + mr;
  if (arow >= M) arow = M - 1;  // clamp: keeps EXEC all-1s for WMMA
  const float scale = rowScale ? rowScale[(size_t)arow * NREL] : 1.0f;
  const float* Ap = A + (size_t)arow * K + half * 2;

  v8f acc = {};
  for (int k = 0; k < K; k += 4) {
    v2f a;
    a.x = Ap[k] * scale;        // A[arow][k + half*2]
    a.y = Ap[k + 1] * scale;    // A[arow][k + half*2 + 1]
    const int kk = k + half * 2;
    v2f b;
    if (TRANSB) {
      const float* Bp = B + (size_t)(nbase + mr) * K + kk;
      b.x = Bp[0];
      b.y = Bp[1];
    } else {
      b.x = B[(size_t)kk * DCH + nbase + mr];
      b.y = B[(size_t)(kk + 1) * DCH + nbase + mr];
    }
    acc = __builtin_amdgcn_wmma_f32_16x16x4_f32(
        /*neg_a=*/false, a, /*neg_b=*/false, b,
        /*c_mod=*/(short)0, acc, /*reuse_a=*/false, /*reuse_b=*/false);
  }

  const int col = nbase + mr;
#pragma unroll
  for (int j = 0; j < 8; ++j) {
    const int row = mbase + j + half * 8;  // C layout: VGPR j -> M=j / M=j+8
    if (row < M) {
      float v = acc[j];
      if (ACCUM)
        C[(size_t)row * DCH + col] += v;
      else
        C[(size_t)row * DCH + col] = v + (bias ? bias[col] : 0.0f);
    }
  }
}

// ------------------------------------------------------------ LayerNorm
// One wave32 per row of 128; float4 per lane; shfl_xor tree reductions.
__global__ void layernorm_kernel(const float* __restrict__ x,
                                 const float* __restrict__ gamma,
                                 const float* __restrict__ beta,
                                 float* __restrict__ out, int N) {
  int warp = (blockIdx.x * blockDim.x + threadIdx.x) >> 5;
  int lane = threadIdx.x & 31;
  if (warp >= N) return;
  const float4 v = ((const float4*)(x + (size_t)warp * DCH))[lane];
  float s = v.x + v.y + v.z + v.w;
#pragma unroll
  for (int off = 16; off > 0; off >>= 1) s += __shfl_xor(s, off, 32);
  const float mu = s * (1.0f / DCH);
  const float dx = v.x - mu, dy = v.y - mu, dz = v.z - mu, dw = v.w - mu;
  float ss = dx * dx + dy * dy + dz * dz + dw * dw;
#pragma unroll
  for (int off = 16; off > 0; off >>= 1) ss += __shfl_xor(ss, off, 32);
  const float rs = rsqrtf(ss * (1.0f / DCH) + 1e-5f);
  const float4 g = ((const float4*)gamma)[lane];
  const float4 bb = ((const float4*)beta)[lane];
  float4 o;
  o.x = dx * rs * g.x + bb.x;
  o.y = dy * rs * g.y + bb.y;
  o.z = dz * rs * g.z + bb.z;
  o.w = dw * rs * g.w + bb.w;
  ((float4*)(out + (size_t)warp * DCH))[lane] = o;
}

// ---------------------------------------------------------------- driver

extern "C" void kernel_launch(void* const* d_in, const int* in_sizes, int n_in,
                              void* d_out, int out_size, void* d_ws, size_t ws_size,
                              hipStream_t stream) {
  const float* x0 = (const float*)d_in[0];
  const float* x1 = (const float*)d_in[1];
  const int* edge_index = (const int*)d_in[2];
  const int* edge_type = (const int*)d_in[3];
  const float* lin0_w = (const float*)d_in[4];
  const float* lin0_b = (const float*)d_in[5];
  const float* lin1_w = (const float*)d_in[6];
  const float* lin1_b = (const float*)d_in[7];
  const float* comp[3] = {(const float*)d_in[8], (const float*)d_in[12], (const float*)d_in[16]};
  const float* bases[3] = {(const float*)d_in[9], (const float*)d_in[13], (const float*)d_in[17]};
  const float* root[3] = {(const float*)d_in[10], (const float*)d_in[14], (const float*)d_in[18]};
  const float* cbias[3] = {(const float*)d_in[11], (const float*)d_in[15], (const float*)d_in[19]};
  const float* gamma = (const float*)d_in[20];
  const float* beta = (const float*)d_in[21];

  const int N0 = in_sizes[0] / DCH;        // 25000
  const int K1 = in_sizes[6] / DCH;        // 256 (lin1 input width)
  const int N1 = in_sizes[1] / K1;         // 25000
  const int N = N0 + N1;                   // 50000
  const int E = in_sizes[3];               // 600000

  // Workspace layout (floats): xA | xB | agg | invcnt | Wc[3][4][128][128]
  float* ws = (float*)d_ws;
  float* xA = ws;
  float* xB = xA + (size_t)N * DCH;
  float* agg = xB + (size_t)N * DCH;
  float* invcnt = agg + (size_t)N * DCH;
  float* Wc = invcnt + (size_t)N * NREL;

  const int B256 = 256;
  const dim3 blk(B256);

  // 1) Basis-combined relation weights for all 3 convs.
  for (int c = 0; c < 3; ++c)
    combine_weights_kernel<<<(NREL * DCH * DCH) / B256, blk, 0, stream>>>(
        comp[c], bases[c], Wc + (size_t)c * NREL * DCH * DCH);

  // 2) Per-(dst,rel) inverse mean counts (edge list is shared by all convs).
  const int ncnt = N * NREL;
  fill_kernel<<<(ncnt + B256 - 1) / B256, blk, 0, stream>>>(invcnt, ncnt, 0.0f);
  count_kernel<<<(E + B256 - 1) / B256, blk, 0, stream>>>(edge_index, edge_type, invcnt, E);
  invcnt_kernel<<<(ncnt + B256 - 1) / B256, blk, 0, stream>>>(invcnt, ncnt);

  // 3) Input projections: xA = concat(x0 @ lin0_w^T + b0, x1 @ lin1_w^T + b1)
  wmma_gemm_kernel<true, false><<<(N0 + 15) / 16, blk, 0, stream>>>(
      x0, lin0_w, xA, lin0_b, nullptr, N0, DCH);
  wmma_gemm_kernel<true, false><<<(N1 + 15) / 16, blk, 0, stream>>>(
      x1, lin1_w, xA + (size_t)N0 * DCH, lin1_b, nullptr, N1, K1);

  // 4) Three RGCN convs (aggregate-then-transform, mean folded into GEMM A-scale)
  float* cur = xA;
  float* nxt = xB;
  const int gM = (N + 15) / 16;
  const int nElems = N * DCH;
  for (int c = 0; c < 3; ++c) {
    // self/root transform: nxt = cur @ root + bias
    wmma_gemm_kernel<false, false><<<gM, blk, 0, stream>>>(
        cur, root[c], nxt, cbias[c], nullptr, N, DCH);
    for (int r = 0; r < NREL; ++r) {
      fill_kernel<<<(nElems + B256 - 1) / B256, blk, 0, stream>>>(agg, nElems, 0.0f);
      aggregate_kernel<<<(E * 32 + B256 - 1) / B256, blk, 0, stream>>>(
          cur, edge_index, edge_type, r, agg, E);
      // nxt += (invcnt[:,r] * agg) @ W[c][r]
      wmma_gemm_kernel<false, true><<<gM, blk, 0, stream>>>(
          agg, Wc + ((size_t)c * NREL + r) * DCH * DCH, nxt, nullptr,
          invcnt + r, N, DCH);
    }
    if (c < 2) relu_kernel<<<(nElems + B256 - 1) / B256, blk, 0, stream>>>(nxt, nElems);
    float* t = cur; cur = nxt; nxt = t;
  }

  // 5) LayerNorm of final conv output -> d_out
  layernorm_kernel<<<(N * 32 + B256 - 1) / B256, blk, 0, stream>>>(
      cur, gamma, beta, (float*)d_out, N);
}